// BidiGATv2Conv_34866544509287
// MI455X (gfx1250) — compile-verified
//
#include <hip/hip_runtime.h>
#include <hip/hip_bf16.h>

#define NN 100000
#define EE 1600000
#define DD 64
#define CC 64
#define DE 32
#define NEG_SLOPE 0.2f
#define EPSV 1e-16f

typedef __attribute__((ext_vector_type(2))) float v2f;
typedef __attribute__((ext_vector_type(8))) float v8f;

// ---------------------------------------------------------------------------
// Order-preserving float <-> uint mapping for atomic segment-max
// ---------------------------------------------------------------------------
__device__ __forceinline__ unsigned f2ord(float f) {
    unsigned u = __float_as_uint(f);
    return (u & 0x80000000u) ? ~u : (u | 0x80000000u);
}
__device__ __forceinline__ float ord2f(unsigned u) {
    u = (u & 0x80000000u) ? (u ^ 0x80000000u) : ~u;
    return __uint_as_float(u);
}

// ---------------------------------------------------------------------------
// Node projection: Y[nrows,64] = X[nrows,64] @ W[64,64] via V_WMMA_F32_16X16X4_F32
// One wave (32 lanes) computes a 16x64 output tile; 4 waves per block.
// ---------------------------------------------------------------------------
__global__ __launch_bounds__(128) void proj64_wmma(
    const float* __restrict__ X, const float* __restrict__ W,
    float* __restrict__ Y, int nrows) {
    __shared__ float sW[64 * 64];
    __shared__ float sA[4][16 * 65];   // pitch 65 to avoid bank conflicts on A-frag reads

    const int wave = threadIdx.x >> 5;
    const int lane = threadIdx.x & 31;
    const int half = lane >> 4;        // 0: lanes 0-15, 1: lanes 16-31
    const int l15  = lane & 15;

    // Stage W (64x64) cooperatively
    for (int i = threadIdx.x; i < 64 * 64; i += 128) sW[i] = W[i];

    const int tile = blockIdx.x * 4 + wave;
    const int row0 = tile * 16;

    // Stage this wave's 16x64 A tile (zero-fill OOB rows)
    for (int i = lane; i < 16 * 64; i += 32) {
        int r = i >> 6, c = i & 63;
        int gr = row0 + r;
        sA[wave][r * 65 + c] = (gr < nrows) ? X[(size_t)gr * 64 + c] : 0.0f;
    }
    __syncthreads();

    if (row0 < nrows) {
        v8f acc[4] = {};
        #pragma unroll
        for (int t = 0; t < 16; ++t) {
            const int k0 = t * 4;
            // A fragment (16x4 f32): VGPR0 = {K=k0 | K=k0+2}, VGPR1 = {K=k0+1 | K=k0+3}
            v2f a;
            a.x = sA[wave][l15 * 65 + k0 + 2 * half + 0];
            a.y = sA[wave][l15 * 65 + k0 + 2 * half + 1];
            #pragma unroll
            for (int nb = 0; nb < 4; ++nb) {
                // B fragment (4x16 f32), same K split across lane halves
                v2f b;
                b.x = sW[(k0 + 2 * half + 0) * 64 + nb * 16 + l15];
                b.y = sW[(k0 + 2 * half + 1) * 64 + nb * 16 + l15];
                acc[nb] = __builtin_amdgcn_wmma_f32_16x16x4_f32(
                    false, a, false, b, (short)0, acc[nb], false, false);
            }
        }
        // C/D layout: VGPR j -> rows {j, j+8} across lane halves, N = l15
        #pragma unroll
        for (int nb = 0; nb < 4; ++nb) {
            #pragma unroll
            for (int j = 0; j < 8; ++j) {
                int r = row0 + j + 8 * half;
                if (r < nrows) Y[(size_t)r * 64 + nb * 16 + l15] = acc[nb][j];
            }
        }
    }
}

// ---------------------------------------------------------------------------
// Init: outputs = bias broadcast; max-keys = 0 (== -inf); denominators = 0
// ---------------------------------------------------------------------------
__global__ void init_kernel(float* __restrict__ out_b, float* __restrict__ out_f,
                            const float* __restrict__ b_b, const float* __restrict__ b_f,
                            unsigned* __restrict__ mxk_b, unsigned* __restrict__ mxk_f,
                            float* __restrict__ den_b, float* __restrict__ den_f) {
    int i = blockIdx.x * blockDim.x + threadIdx.x;
    if (i < NN * 64) {
        out_b[i] = b_b[i & 63];
        out_f[i] = b_f[i & 63];
    }
    if (i < NN) {
        mxk_b[i] = 0u; mxk_f[i] = 0u;
        den_b[i] = 0.0f; den_f[i] = 0.0f;
    }
}

// ---------------------------------------------------------------------------
// Pass 1: per-edge logit = att . leaky_relu(xl[src] + xr[dst] + e_attr@We)
// One wave per edge; lane L owns channels {L, L+32}. Atomic segment-max on dst.
// ---------------------------------------------------------------------------
__global__ __launch_bounds__(256) void edge_logits_kernel(
    const float* __restrict__ xl, const float* __restrict__ xr,
    const long long* __restrict__ src, const long long* __restrict__ dst,
    const float* __restrict__ eattr, const float* __restrict__ We,
    const float* __restrict__ att,
    float* __restrict__ logits, unsigned* __restrict__ mxkey) {
    __shared__ float sWe[32 * 64];
    __shared__ float sAtt[64];
    for (int i = threadIdx.x; i < 32 * 64; i += 256) sWe[i] = We[i];
    if (threadIdx.x < 64) sAtt[threadIdx.x] = att[threadIdx.x];
    __syncthreads();

    const int wave = threadIdx.x >> 5;
    const int lane = threadIdx.x & 31;
    const int e = blockIdx.x * 8 + wave;
    if (e >= EE) return;

    const long long s = src[e];
    const long long d = dst[e];

    const float ea = eattr[(size_t)e * DE + lane];   // lane k holds e_attr[e][k]
    float acc0 = 0.0f, acc1 = 0.0f;
    #pragma unroll
    for (int k = 0; k < 32; ++k) {
        float eak = __shfl(ea, k, 32);
        acc0 += eak * sWe[k * 64 + lane];
        acc1 += eak * sWe[k * 64 + lane + 32];
    }
    float m0 = xl[(size_t)s * 64 + lane]      + xr[(size_t)d * 64 + lane]      + acc0;
    float m1 = xl[(size_t)s * 64 + lane + 32] + xr[(size_t)d * 64 + lane + 32] + acc1;
    m0 = (m0 > 0.0f) ? m0 : m0 * NEG_SLOPE;
    m1 = (m1 > 0.0f) ? m1 : m1 * NEG_SLOPE;
    float p = m0 * sAtt[lane] + m1 * sAtt[lane + 32];
    #pragma unroll
    for (int off = 16; off > 0; off >>= 1) p += __shfl_down(p, off, 32);

    if (lane == 0) {
        logits[e] = p;
        atomicMax(&mxkey[d], f2ord(p));
    }
}

// ---------------------------------------------------------------------------
// Pass 2: ex = exp(logit - mx[dst]); denom[dst] += ex  (ex overwrites logits)
// ---------------------------------------------------------------------------
__global__ void edge_exp_kernel(const long long* __restrict__ dst,
                                const unsigned* __restrict__ mxkey,
                                float* __restrict__ logits,
                                float* __restrict__ denom) {
    int e = blockIdx.x * blockDim.x + threadIdx.x;
    if (e >= EE) return;
    const long long d = dst[e];
    float ex = __expf(logits[e] - ord2f(mxkey[d]));   // <= 1, no overflow
    logits[e] = ex;
    atomicAdd(&denom[d], ex);
}

// ---------------------------------------------------------------------------
// Pass 3: out[dst] += (ex / (denom[dst]+eps)) * xl[src]; thread per (edge,chan)
// ---------------------------------------------------------------------------
__global__ void edge_scatter_kernel(const long long* __restrict__ src,
                                    const long long* __restrict__ dst,
                                    const float* __restrict__ ex,
                                    const float* __restrict__ denom,
                                    const float* __restrict__ xl,
                                    float* __restrict__ out) {
    long long idx = (long long)blockIdx.x * blockDim.x + threadIdx.x;
    int e = (int)(idx >> 6);
    int c = (int)(idx & 63);
    if (e >= EE) return;
    const long long s = src[e];
    const long long d = dst[e];
    float alpha = ex[e] / (denom[d] + EPSV);
    atomicAdd(&out[(size_t)d * 64 + c], alpha * xl[(size_t)s * 64 + c]);
}

// ---------------------------------------------------------------------------
extern "C" void kernel_launch(void* const* d_in, const int* in_sizes, int n_in,
                              void* d_out, int out_size, void* d_ws, size_t ws_size,
                              hipStream_t stream) {
    const float*      x0   = (const float*)d_in[0];
    const float*      x1   = (const float*)d_in[1];
    const long long*  ei   = (const long long*)d_in[2];   // int64 per reference
    const float*      ea   = (const float*)d_in[3];
    const float*      Wl_b = (const float*)d_in[4];
    const float*      Wr_b = (const float*)d_in[5];
    const float*      We_b = (const float*)d_in[6];
    const float*      at_b = (const float*)d_in[7];
    const float*      b_b  = (const float*)d_in[8];
    const float*      Wl_f = (const float*)d_in[9];
    const float*      Wr_f = (const float*)d_in[10];
    const float*      We_f = (const float*)d_in[11];
    const float*      at_f = (const float*)d_in[12];
    const float*      b_f  = (const float*)d_in[13];

    const long long* srcI = ei;        // edge_index[0]
    const long long* dstI = ei + EE;   // edge_index[1]

    float* out_b = (float*)d_out;              // [N,64]
    float* out_f = (float*)d_out + (size_t)NN * 64;

    // Workspace carve-up (all fp32-sized slots)
    float* ws = (float*)d_ws;
    const size_t N64 = (size_t)NN * 64;
    float*    xl_b   = ws;                 ws += N64;
    float*    xr_b   = ws;                 ws += N64;
    float*    xl_f   = ws;                 ws += N64;
    float*    xr_f   = ws;                 ws += N64;
    float*    lg_b   = ws;                 ws += EE;
    float*    lg_f   = ws;                 ws += EE;
    unsigned* mxk_b  = (unsigned*)ws;      ws += NN;
    unsigned* mxk_f  = (unsigned*)ws;      ws += NN;
    float*    den_b  = ws;                 ws += NN;
    float*    den_f  = ws;                 ws += NN;
    (void)ws_size; (void)in_sizes; (void)n_in; (void)out_size;

    // 0) init outputs (bias) + segment accumulators
    {
        int total = NN * 64;
        init_kernel<<<(total + 255) / 256, 256, 0, stream>>>(
            out_b, out_f, b_b, b_f, mxk_b, mxk_f, den_b, den_f);
    }

    // 1) four node projections (WMMA fp32); L2-resident afterwards (102 MB < 192 MB)
    {
        int tiles  = (NN + 15) / 16;
        int blocks = (tiles + 3) / 4;
        proj64_wmma<<<blocks, 128, 0, stream>>>(x0, Wl_b, xl_b, NN);
        proj64_wmma<<<blocks, 128, 0, stream>>>(x1, Wr_b, xr_b, NN);
        proj64_wmma<<<blocks, 128, 0, stream>>>(x1, Wl_f, xl_f, NN);
        proj64_wmma<<<blocks, 128, 0, stream>>>(x0, Wr_f, xr_f, NN);
    }

    // 2) edge logits + segment max (backward: dst=ei[1]; forward: dst=ei[0])
    {
        int blocks = (EE + 7) / 8;   // 8 edges (waves) per 256-thread block
        edge_logits_kernel<<<blocks, 256, 0, stream>>>(
            xl_b, xr_b, srcI, dstI, ea, We_b, at_b, lg_b, mxk_b);
        edge_logits_kernel<<<blocks, 256, 0, stream>>>(
            xl_f, xr_f, dstI, srcI, ea, We_f, at_f, lg_f, mxk_f);
    }

    // 3) exp + segment sum
    {
        int blocks = (EE + 255) / 256;
        edge_exp_kernel<<<blocks, 256, 0, stream>>>(dstI, mxk_b, lg_b, den_b);
        edge_exp_kernel<<<blocks, 256, 0, stream>>>(srcI, mxk_f, lg_f, den_f);
    }

    // 4) normalized scatter-add of xl[src] onto dst
    {
        long long total = (long long)EE * 64;
        int blocks = (int)((total + 255) / 256);
        edge_scatter_kernel<<<blocks, 256, 0, stream>>>(srcI, dstI, lg_b, den_b, xl_b, out_b);
        edge_scatter_kernel<<<blocks, 256, 0, stream>>>(dstI, srcI, lg_f, den_f, xl_f, out_f);
    }
}